// BoxBlur_42167988912512
// MI455X (gfx1250) — compile-verified
//
#include <hip/hip_runtime.h>
#include <stdint.h>

typedef float v2f __attribute__((ext_vector_type(2)));
typedef float v8f __attribute__((ext_vector_type(8)));

#define IMG_H 512
#define IMG_W 512
#define TILE_R 32
#define TILE_C 64
#define IN_RS (TILE_R + 6)   // 38 rows incl. vertical halo
#define IN_CS (TILE_C + 6)   // 70 cols incl. horizontal halo
#define IN_PITCH 72          // padded LDS pitch (bank-conflict avoidance)
#define H_RS (IN_RS + 2)     // 40: +2 zero rows so K=22,23 pad reads are real zeros
#define H_PITCH 68
#define INV49 (1.0f / 49.0f)

__device__ __forceinline__ int refl(int i, int n) {
    // jnp.pad mode="reflect": -1 -> 1, n -> n-2 (edge not repeated)
    i = (i < 0) ? -i : i;
    i = (i >= n) ? (2 * n - 2 - i) : i;
    return i;
}

__global__ __launch_bounds__(256)
void boxblur7_sep_wmma(const float* __restrict__ x, float* __restrict__ out) {
    __shared__ float sIn[IN_RS * IN_PITCH];  // input tile (with halo)
    __shared__ float sH [H_RS  * H_PITCH];   // horizontal 7-tap raw sums + 2 pad rows

    const int plane = blockIdx.z;                  // (n, c) plane, 0..95
    const int R0 = blockIdx.y * TILE_R;
    const int C0 = blockIdx.x * TILE_C;
    const float* __restrict__ xp = x   + (size_t)plane * (IMG_H * IMG_W);
    float* __restrict__       op = out + (size_t)plane * (IMG_H * IMG_W);

    const int tid = threadIdx.x;

    // ---- Stage 1: async global->LDS copy of 38x70 input tile (reflect borders).
    // GLOBAL_LOAD_ASYNC_TO_LDS_B32: per-lane global address, per-lane LDS offset,
    // tracked by ASYNCcnt -> no VGPR staging, whole tile fetch is in flight at once.
    for (int idx = tid; idx < IN_RS * IN_CS; idx += 256) {
        const int r = idx / IN_CS;
        const int c = idx - r * IN_CS;
        const int gr = refl(R0 - 3 + r, IMG_H);
        const int gc = refl(C0 - 3 + c, IMG_W);
        const float* gaddr = xp + (gr * IMG_W + gc);
        uint32_t ldsoff = (uint32_t)(uintptr_t)
            ((__attribute__((address_space(3))) float*)&sIn[r * IN_PITCH + c]);
        asm volatile("global_load_async_to_lds_b32 %0, %1, off"
                     :: "v"(ldsoff), "v"(gaddr) : "memory");
    }
    asm volatile("s_wait_asynccnt 0x0" ::: "memory");
    __syncthreads();

    // ---- Stage 2: horizontal raw 7-tap sums (VALU; cheap, memory-bound kernel) ----
    for (int idx = tid; idx < IN_RS * TILE_C; idx += 256) {
        const int r = idx >> 6;       // TILE_C == 64
        const int c = idx & 63;
        const float* p = &sIn[r * IN_PITCH + c];
        sH[r * H_PITCH + c] = ((p[0] + p[1]) + (p[2] + p[3])) +
                              ((p[4] + p[5]) + p[6]);
    }
    // zero the two K-padding rows (38, 39): keeps all WMMA B loads unconditional
    if (tid < 2 * TILE_C) {
        sH[(IN_RS + (tid >> 6)) * H_PITCH + (tid & 63)] = 0.0f;
    }
    __syncthreads();

    // ---- Stage 3: vertical pass on the matrix core ----
    // V(16x16) = A(16x24) x B(24x16); A[m][t] = 1/49 iff (unsigned)(t-m) <= 6
    // (naturally 0 for pad rows t=22,23), B[t][n] = sH[tile_row + t][tile_col + n].
    // Six chained V_WMMA_F32_16X16X4_F32 per wave, full EXEC, no divergence.
    const int wave = tid >> 5;
    const int lane = tid & 31;
    const int half = lane >> 4;   // which 16-lane half
    const int l16  = lane & 15;
    const int tr = (wave >> 2) * 16;   // 0 / 16 : row sub-tile within block tile
    const int tc = (wave &  3) * 16;   // 0/16/32/48 : col sub-tile

    const float* sHb = &sH[tr * H_PITCH + tc + l16];

    v8f acc = {};
#pragma unroll
    for (int k0 = 0; k0 < 24; k0 += 4) {
        const int t0 = k0 + 2 * half;      // K indices this lane holds (A doc layout,
        const int t1 = t0 + 1;             // B assumed symmetric with A)
        v2f a, b;
        a.x = ((uint32_t)(t0 - l16) <= 6u) ? INV49 : 0.0f;
        a.y = ((uint32_t)(t1 - l16) <= 6u) ? INV49 : 0.0f;
        b.x = sHb[t0 * H_PITCH];
        b.y = sHb[t1 * H_PITCH];
        acc = __builtin_amdgcn_wmma_f32_16x16x4_f32(
            /*neg_a=*/false, a, /*neg_b=*/false, b,
            /*c_mod=*/(short)0, acc, /*reuse_a=*/false, /*reuse_b=*/false);
    }

    // ---- Store: documented C/D layout (VGPR i -> rows i / i+8 per lane half) ----
    const int orow = R0 + tr + 8 * half;
    const int ocol = C0 + tc + l16;
#pragma unroll
    for (int i = 0; i < 8; ++i) {
        op[(size_t)(orow + i) * IMG_W + ocol] = acc[i];
    }
}

extern "C" void kernel_launch(void* const* d_in, const int* in_sizes, int n_in,
                              void* d_out, int out_size, void* d_ws, size_t ws_size,
                              hipStream_t stream) {
    const float* x = (const float*)d_in[0];
    float* out = (float*)d_out;
    const int planes = in_sizes[0] / (IMG_H * IMG_W);   // 32 * 3 = 96
    dim3 grid(IMG_W / TILE_C, IMG_H / TILE_R, planes);
    boxblur7_sep_wmma<<<grid, dim3(256), 0, stream>>>(x, out);
}